// SphericalFourierLayer_15908558864854
// MI455X (gfx1250) — compile-verified
//
#include <hip/hip_runtime.h>
#include <math.h>
#include <stdint.h>

// ---------------------------------------------------------------------------
// SphericalFourierLayer on MI455X (gfx1250), all heavy math as f32 WMMA GEMMs.
// B=8, Cin=Cout=32, H=360, W=720, modes 64x64.
//   A: X1T[bc][n(128: re||im)][h]  = x[bc][h][:720] . FwT  (rDFT over W, ortho)
//   B: X2[bc][2][l][m]             = G[64x360] (x) X1     (DFT over H)
//   C: X3T[bo][2][m][l]            = per-mode complex channel mix (VALU)
//   D: Y1[(bo,h)][128]             = D[360x64] (x) X3     (iDFT over H)
//   E: y[bo][h][w]                 = Y1 . ET               (irDFT over W, x2 fold)
// Tile fills for A/E use GLOBAL_LOAD_ASYNC_TO_LDS_B128 (ASYNCcnt-tracked DMA).
// ---------------------------------------------------------------------------

typedef __attribute__((ext_vector_type(2))) float v2f;
typedef __attribute__((ext_vector_type(8))) float v8f;

__device__ __forceinline__ v8f wmma4(v2f a, v2f b, v8f c) {
  // D = A(16x4) * B(4x16) + C(16x16), full f32
  return __builtin_amdgcn_wmma_f32_16x16x4_f32(false, a, false, b, (short)0, c,
                                               false, false);
}

// async 16B global -> LDS copy (per-lane); tracked with ASYNCcnt
__device__ __forceinline__ void async_g2l_b128(float* lds_dst, const float* g) {
  unsigned loff = (unsigned)(uintptr_t)lds_dst;  // addr[31:0] = LDS offset
  asm volatile("global_load_async_to_lds_b128 %0, %1, off"
               :: "v"(loff), "v"(g) : "memory");
}
__device__ __forceinline__ void async_wait0() {
  asm volatile("s_wait_asynccnt 0x0" ::: "memory");
}

#define TWO_PI 6.283185307179586f

// ---------------- workspace layout (floats) ----------------
constexpr size_t SZ_FWT = 128ull * 720;        // rDFT matrix, transposed [n][w]
constexpr size_t SZ_G   = 64ull * 360;         // H-DFT cos / sin  [l][h]
constexpr size_t SZ_D   = 368ull * 64;         // H-iDFT cos / sin [h(pad 368)][l]
constexpr size_t SZ_ET  = 720ull * 128;        // irDFT matrix, transposed [w][k]
constexpr size_t SZ_X1  = 256ull * 128 * 360;  // X1T (aliased with Y1)
constexpr size_t SZ_X2  = 256ull * 2 * 64 * 64;
constexpr size_t OFF_FWT = 0;
constexpr size_t OFF_GC  = OFF_FWT + SZ_FWT;
constexpr size_t OFF_GS  = OFF_GC + SZ_G;
constexpr size_t OFF_DC  = OFF_GS + SZ_G;
constexpr size_t OFF_DS  = OFF_DC + SZ_D;
constexpr size_t OFF_ET  = OFF_DS + SZ_D;
constexpr size_t OFF_X1  = OFF_ET + SZ_ET;     // also Y1
constexpr size_t OFF_X2  = OFF_X1 + SZ_X1;
constexpr size_t OFF_X3  = OFF_X2 + SZ_X2;

// ---------------- twiddle-matrix init (tiny, rebuilt per call) -------------
__global__ void initFw(float* __restrict__ FwT) {
  int i = blockIdx.x * 256 + threadIdx.x;           // 128*720
  if (i >= 128 * 720) return;
  int n = i / 720, w = i - n * 720, m = n & 63;
  float ang = TWO_PI * (float)((m * w) % 720) / 720.0f;
  float s = rsqrtf(259200.0f);                      // 1/sqrt(H*W) ortho
  FwT[i] = (n < 64) ? cosf(ang) * s : -sinf(ang) * s;
}
__global__ void initG(float* __restrict__ Gc, float* __restrict__ Gs) {
  int i = blockIdx.x * 256 + threadIdx.x;           // 64*360
  if (i >= 64 * 360) return;
  int l = i / 360, h = i - l * 360;
  float ang = TWO_PI * (float)((l * h) % 360) / 360.0f;
  Gc[i] = cosf(ang); Gs[i] = sinf(ang);
}
__global__ void initD(float* __restrict__ Dc, float* __restrict__ Ds) {
  int i = blockIdx.x * 256 + threadIdx.x;           // 368*64
  if (i >= 368 * 64) return;
  int h = i >> 6, l = i & 63;
  if (h >= 360) { Dc[i] = 0.0f; Ds[i] = 0.0f; return; }
  float ang = TWO_PI * (float)((l * h) % 360) / 360.0f;
  Dc[i] = cosf(ang); Ds[i] = sinf(ang);
}
__global__ void initE(float* __restrict__ ET) {
  int i = blockIdx.x * 256 + threadIdx.x;           // 720*128
  if (i >= 720 * 128) return;
  int w = i >> 7, k = i & 127, m = k & 63;
  float ang = TWO_PI * (float)((m * w) % 720) / 720.0f;
  float s = rsqrtf(259200.0f);
  float fac = (m == 0) ? 1.0f : 2.0f;               // Hermitian fold
  ET[i] = (k < 64) ? fac * cosf(ang) * s
                   : ((m == 0) ? 0.0f : -fac * sinf(ang) * s);
}

// ---------------- Stage A: rDFT over W --------------------------------------
// grid.x = 92160/16 ; 8 waves, each owns one N-tile of 16 of the 128 outputs.
#define APITCH 724   // row pitch (floats): 2896 B = 181*16, bank-conflict free
__global__ __launch_bounds__(256) void stageA(const float* __restrict__ x,
                                              const float* __restrict__ FwT,
                                              float* __restrict__ X1T) {
  __shared__ float lds[16 * APITCH];
  const int tid = threadIdx.x;
  const int mbase = blockIdx.x * 16;                // global row = bc*360+h
  // async DMA fill: 16 rows x 720 floats = 2880 x 16B chunks
  for (int i = tid; i < 16 * 180; i += 256) {
    int r = i / 180, c16 = i - r * 180;
    async_g2l_b128(lds + r * APITCH + c16 * 4,
                   x + (size_t)(mbase + r) * 720 + c16 * 4);
  }
  async_wait0();
  __syncthreads();
  const int lane = tid & 31, wave = tid >> 5;
  const int lx = lane & 15, hi = lane >> 4;
  const int n = wave * 16 + lx;                     // output col 0..127
  const float* ap = lds + lx * APITCH + hi * 2;
  const float* bp = FwT + (size_t)n * 720 + hi * 2;
  v8f acc = {};
  for (int k = 0; k < 180; ++k) {
    v2f a = *(const v2f*)(ap + k * 4);
    v2f b = *(const v2f*)(bp + k * 4);
    acc = wmma4(a, b, acc);
  }
#pragma unroll
  for (int r = 0; r < 8; ++r) {                     // store transposed [bc][n][h]
    int row = mbase + r + 8 * hi;
    int bc = row / 360, h = row - bc * 360;
    X1T[((size_t)bc * 128 + n) * 360 + h] = acc[r];
  }
}

// ---------------- Stage B: DFT over H (complex), per bc ---------------------
// grid.x = 256 (bc); 8 waves * 2 tiles = 16 tiles (4 l-tiles x 4 m-tiles)
__global__ __launch_bounds__(256) void stageB(const float* __restrict__ X1T,
                                              const float* __restrict__ Gc,
                                              const float* __restrict__ Gs,
                                              float* __restrict__ X2) {
  const int bc = blockIdx.x;
  const int lane = threadIdx.x & 31, wave = threadIdx.x >> 5;
  const int lx = lane & 15, hi = lane >> 4;
  const float* xb = X1T + (size_t)bc * 128 * 360;
  for (int t = 0; t < 2; ++t) {
    int tile = wave * 2 + t;
    int lt = tile >> 2, mt = tile & 3;
    const float* ac = Gc + (size_t)(lt * 16 + lx) * 360 + hi * 2;
    const float* as = Gs + (size_t)(lt * 16 + lx) * 360 + hi * 2;
    const float* br = xb + (size_t)(mt * 16 + lx) * 360 + hi * 2;
    const float* bi = xb + (size_t)(64 + mt * 16 + lx) * 360 + hi * 2;
    v8f are = {}, aim = {};
    for (int k = 0; k < 90; ++k) {
      v2f c = *(const v2f*)(ac + k * 4);
      v2f s = *(const v2f*)(as + k * 4);
      v2f xr = *(const v2f*)(br + k * 4);
      v2f xi = *(const v2f*)(bi + k * 4);
      v2f sn = -s;
      are = wmma4(c, xr, are);   // re:  Gc*re + Gs*im   (e^{-i}: c - i s)
      are = wmma4(s, xi, are);
      aim = wmma4(c, xi, aim);   // im:  Gc*im - Gs*re
      aim = wmma4(sn, xr, aim);
    }
    float* ore = X2 + ((size_t)bc * 2 + 0) * 4096;
    float* oim = X2 + ((size_t)bc * 2 + 1) * 4096;
#pragma unroll
    for (int r = 0; r < 8; ++r) {
      int l = lt * 16 + r + 8 * hi, m = mt * 16 + lx;
      ore[l * 64 + m] = are[r];
      oim[l * 64 + m] = aim[r];
    }
  }
}

// ---------------- Stage C: per-mode 32->32 complex channel mix (VALU) -------
__global__ __launch_bounds__(256) void stageC(const float* __restrict__ X2,
                                              const float* __restrict__ W,
                                              float* __restrict__ X3T) {
  int idx = blockIdx.x * 256 + threadIdx.x;         // ((b*32+o)*64+l)*64+m
  int m = idx & 63, l = (idx >> 6) & 63, o = (idx >> 12) & 31, b = idx >> 17;
  float ar = 0.0f, ai = 0.0f;
  for (int c = 0; c < 32; ++c) {
    const float* xp = X2 + ((size_t)(b * 32 + c) * 2) * 4096 + l * 64 + m;
    float xr = xp[0], xi = xp[4096];
    v2f wc = *(const v2f*)(W + ((((size_t)c * 32 + o) * 64 + l) * 64 + m) * 2);
    ar += xr * wc.x - xi * wc.y;
    ai += xr * wc.y + xi * wc.x;
  }
  float* out = X3T + ((size_t)(b * 32 + o) * 2) * 4096;   // transposed [m][l]
  out[m * 64 + l] = ar;
  out[4096 + m * 64 + l] = ai;
}

// ---------------- Stage D: iDFT over H (complex), per bo --------------------
// 256 bo * (23 h-tiles * 4 m-tiles = 92 tiles, padded to 96); 8 waves/WG.
__global__ __launch_bounds__(256) void stageD(const float* __restrict__ X3T,
                                              const float* __restrict__ Dc,
                                              const float* __restrict__ Ds,
                                              float* __restrict__ Y1) {
  const int lane = threadIdx.x & 31, wave = threadIdx.x >> 5;
  const int wg = blockIdx.x * 8 + wave;             // [0, 24576)
  const int bo = wg / 96, tl = wg - bo * 96;
  if (tl >= 92) return;
  const int ht = tl >> 2, mt = tl & 3;
  const int lx = lane & 15, hi = lane >> 4;
  const float* ac = Dc + (size_t)(ht * 16 + lx) * 64 + hi * 2;
  const float* as = Ds + (size_t)(ht * 16 + lx) * 64 + hi * 2;
  const float* br = X3T + ((size_t)bo * 2 + 0) * 4096 + (mt * 16 + lx) * 64 + hi * 2;
  const float* bi = X3T + ((size_t)bo * 2 + 1) * 4096 + (mt * 16 + lx) * 64 + hi * 2;
  v8f are = {}, aim = {};
  for (int k = 0; k < 16; ++k) {
    v2f c = *(const v2f*)(ac + k * 4);
    v2f s = *(const v2f*)(as + k * 4);
    v2f xr = *(const v2f*)(br + k * 4);
    v2f xi = *(const v2f*)(bi + k * 4);
    v2f sn = -s;
    are = wmma4(c, xr, are);   // re: Dc*re - Ds*im    (e^{+i}: c + i s)
    are = wmma4(sn, xi, are);
    aim = wmma4(s, xr, aim);   // im: Ds*re + Dc*im
    aim = wmma4(c, xi, aim);
  }
  const int m = mt * 16 + lx;
#pragma unroll
  for (int r = 0; r < 8; ++r) {
    int h = ht * 16 + r + 8 * hi;
    if (h < 360) {
      float* row = Y1 + ((size_t)bo * 360 + h) * 128;
      row[m] = are[r];
      row[64 + m] = aim[r];
    }
  }
}

// ---------------- Stage E: irDFT over W -------------------------------------
// grid (92160/16, 6); 8 waves each own one of 45 w-tiles per chunk.
#define EPITCH 132   // 528 B = 33*16, bank-conflict free
__global__ __launch_bounds__(256) void stageE(const float* __restrict__ Y1,
                                              const float* __restrict__ ET,
                                              float* __restrict__ y) {
  __shared__ float lds[16 * EPITCH];
  const int tid = threadIdx.x;
  const int mbase = blockIdx.x * 16;
  // async DMA fill: 16 rows x 128 floats = 512 x 16B chunks (2 iters/thread)
  for (int i = tid; i < 16 * 32; i += 256) {
    int r = i >> 5, c16 = i & 31;
    async_g2l_b128(lds + r * EPITCH + c16 * 4,
                   Y1 + (size_t)(mbase + r) * 128 + c16 * 4);
  }
  async_wait0();
  __syncthreads();
  const int lane = tid & 31, wave = tid >> 5;
  const int lx = lane & 15, hi = lane >> 4;
  const int ntile = blockIdx.y * 8 + wave;
  if (ntile >= 45) return;
  const int n = ntile * 16 + lx;                    // w column
  const float* ap = lds + lx * EPITCH + hi * 2;
  const float* bp = ET + (size_t)n * 128 + hi * 2;
  v8f acc = {};
  for (int k = 0; k < 32; ++k) {
    v2f a = *(const v2f*)(ap + k * 4);
    v2f b = *(const v2f*)(bp + k * 4);
    acc = wmma4(a, b, acc);
  }
#pragma unroll
  for (int r = 0; r < 8; ++r)
    y[(size_t)(mbase + r + 8 * hi) * 720 + n] = acc[r];
}

// ---------------------------------------------------------------------------
extern "C" void kernel_launch(void* const* d_in, const int* in_sizes, int n_in,
                              void* d_out, int out_size, void* d_ws, size_t ws_size,
                              hipStream_t stream) {
  const float* x = (const float*)d_in[0];
  const float* w = (const float*)d_in[1];
  float* y = (float*)d_out;
  float* ws = (float*)d_ws;

  float* FwT = ws + OFF_FWT;
  float* Gc  = ws + OFF_GC;
  float* Gs  = ws + OFF_GS;
  float* Dc  = ws + OFF_DC;
  float* Ds  = ws + OFF_DS;
  float* ET  = ws + OFF_ET;
  float* X1T = ws + OFF_X1;   // stage A out / stage B in
  float* X2  = ws + OFF_X2;
  float* X3T = ws + OFF_X3;
  float* Y1  = ws + OFF_X1;   // alias: X1T dead after stage B

  initFw<<<360, 256, 0, stream>>>(FwT);
  initG <<<90,  256, 0, stream>>>(Gc, Gs);
  initD <<<92,  256, 0, stream>>>(Dc, Ds);
  initE <<<360, 256, 0, stream>>>(ET);

  stageA<<<5760, 256, 0, stream>>>(x, FwT, X1T);
  stageB<<<256,  256, 0, stream>>>(X1T, Gc, Gs, X2);
  stageC<<<4096, 256, 0, stream>>>(X2, w, X3T);
  stageD<<<3072, 256, 0, stream>>>(X3T, Dc, Ds, Y1);
  stageE<<<dim3(5760, 6), 256, 0, stream>>>(Y1, ET, y);
}